// FFFWrapper_78692390797431
// MI455X (gfx1250) — compile-verified
//
#include <hip/hip_runtime.h>
#include <hip/hip_bf16.h>

#define DEPTH 11
#define IN_W 768
#define LEAF_W 16
#define OUT_W 768
#define N_NODES 2047
#define N_LEAVES 2048
#define BATCH 8192

typedef __attribute__((ext_vector_type(2))) float v2f;
typedef __attribute__((ext_vector_type(8))) float v8f;

// ---------------------------------------------------------------------------
// Kernel 0: zero the per-leaf counters
// ---------------------------------------------------------------------------
__global__ void fff_zero(int* __restrict__ counts) {
    int i = blockIdx.x * blockDim.x + threadIdx.x;
    if (i < N_LEAVES) counts[i] = 0;
}

// ---------------------------------------------------------------------------
// Kernel 1: tree routing. One wave32 per sample; x row cached in 24 VGPRs,
// node index is wave-uniform so all node-weight loads are fully coalesced.
// ---------------------------------------------------------------------------
__global__ __launch_bounds__(256) void fff_route(
    const float* __restrict__ x,
    const float* __restrict__ node_w,
    const float* __restrict__ node_b,
    int* __restrict__ leaf,
    int* __restrict__ counts)
{
    const int wave = (blockIdx.x * blockDim.x + threadIdx.x) >> 5;
    const int lane = threadIdx.x & 31;
    if (wave >= BATCH) return;

    const float* xr = x + (size_t)wave * IN_W;
    float xs[IN_W / 32];
#pragma unroll
    for (int j = 0; j < IN_W / 32; ++j) xs[j] = xr[lane + 32 * j];

    int node = 0;
#pragma unroll 1
    for (int d = 0; d < DEPTH; ++d) {
        const float* w = node_w + (size_t)node * IN_W;
        float s = 0.f;
#pragma unroll
        for (int j = 0; j < IN_W / 32; ++j) s += xs[j] * w[lane + 32 * j];
        // wave32 butterfly reduction -> every lane holds the full sum
#pragma unroll
        for (int m = 16; m >= 1; m >>= 1) s += __shfl_xor(s, m, 32);
        s += node_b[node];
        node = 2 * node + 1 + (s >= 0.f ? 1 : 0);   // heap child step
    }
    int lf = node - N_NODES;
    if (lane == 0) {
        leaf[wave] = lf;
        atomicAdd(&counts[lf], 1);
    }
}

// ---------------------------------------------------------------------------
// Kernel 2: exclusive prefix scan over 2048 leaf counts (tiny, one block)
// ---------------------------------------------------------------------------
__global__ void fff_scan(const int* __restrict__ counts,
                         int* __restrict__ offsets,
                         int* __restrict__ cursors)
{
    __shared__ int sc[N_LEAVES];
    for (int i = threadIdx.x; i < N_LEAVES; i += blockDim.x) sc[i] = counts[i];
    __syncthreads();
    if (threadIdx.x == 0) {
        int acc = 0;
        for (int i = 0; i < N_LEAVES; ++i) { int c = sc[i]; sc[i] = acc; acc += c; }
    }
    __syncthreads();
    for (int i = threadIdx.x; i < N_LEAVES; i += blockDim.x) {
        offsets[i] = sc[i];
        cursors[i] = sc[i];
    }
}

// ---------------------------------------------------------------------------
// Kernel 3: scatter sample ids into leaf-ordered buckets
// ---------------------------------------------------------------------------
__global__ void fff_scatter(const int* __restrict__ leaf,
                            int* __restrict__ cursors,
                            int* __restrict__ order)
{
    int b = blockIdx.x * blockDim.x + threadIdx.x;
    if (b < BATCH) {
        int pos = atomicAdd(&cursors[leaf[b]], 1);
        order[pos] = b;
    }
}

// ---------------------------------------------------------------------------
// Kernel 4: per-leaf 2-layer MLP using V_WMMA_F32_16X16X4_F32.
// One wave32 per leaf; samples routed to the leaf are processed in 16-row
// tiles so the leaf's 96KB of weights are streamed once per 16-row tile.
//
// Fragment layouts (ISA 7.12.2, fp32):
//   A (16x4):  lane%16 = M row; lanes 0-15 carry K={k0,k0+1} in v[0],v[1],
//              lanes 16-31 carry K={k0+2,k0+3}.
//   B (4x16):  lane%16 = N col; lanes 0-15 carry rows K={k0,k0+1},
//              lanes 16-31 carry rows K={k0+2,k0+3}.
//   C/D (16x16): VGPR r holds M = r + (lane>=16 ? 8 : 0), N = lane%16.
// ---------------------------------------------------------------------------
__global__ __launch_bounds__(32) void fff_leaf_mlp(
    const float* __restrict__ x,
    const float* __restrict__ w1s,
    const float* __restrict__ b1s,
    const float* __restrict__ w2s,
    const int*   __restrict__ order,
    const int*   __restrict__ offsets,
    const int*   __restrict__ counts,
    float* __restrict__ out)
{
    const int lf = blockIdx.x;
    const int cnt = counts[lf];
    if (cnt == 0) return;
    const int base = offsets[lf];

    const float* w1 = w1s + (size_t)lf * IN_W * LEAF_W;   // [768][16] row-major
    const float* w2 = w2s + (size_t)lf * LEAF_W * OUT_W;  // [16][768] row-major

    const int lane  = threadIdx.x;
    const int mn    = lane & 15;          // M (for A) / N (for B,C,D)
    const int koff  = (lane >> 4) * 2;    // 0 for lanes 0-15, 2 for lanes 16-31
    const int rbase = (lane >> 4) * 8;    // C/D row base for this half-wave

    const float bias = b1s[lf * LEAF_W + mn];

    __shared__ float hbuf[16 * 16];       // relu(H) tile, for A-refragmenting

    for (int t0 = 0; t0 < cnt; t0 += 16) {
        const int nrows = min(16, cnt - t0);

        // A-row sample for this lane (clamp padded rows to a valid sample;
        // their results are never stored)
        const int arow = mn;
        const int sidx = order[base + t0 + (arow < nrows ? arow : 0)];
        const float* xrow = x + (size_t)sidx * IN_W;

        // ---- GEMM1: H = Xtile(16x768) * W1(768x16), K-steps of 4 ----
        v8f c = {};
#pragma unroll 4
        for (int k0 = 0; k0 < IN_W; k0 += 4) {
            v2f a, b;
            a[0] = xrow[k0 + koff];
            a[1] = xrow[k0 + koff + 1];
            b[0] = w1[(k0 + koff) * LEAF_W + mn];
            b[1] = w1[(k0 + koff + 1) * LEAF_W + mn];
            c = __builtin_amdgcn_wmma_f32_16x16x4_f32(
                    false, a, false, b, (short)0, c, false, false);
        }

        // bias + relu, spill H through LDS to rebuild A fragments
#pragma unroll
        for (int r = 0; r < 8; ++r) {
            float v = c[r] + bias;
            v = v > 0.f ? v : 0.f;
            hbuf[(rbase + r) * 16 + mn] = v;
        }
        __syncthreads();

        // ---- GEMM2: O = relu(H)(16x16) * W2(16x768), 48 N-tiles ----
#pragma unroll 1
        for (int nt = 0; nt < OUT_W / 16; ++nt) {
            v8f o = {};
#pragma unroll
            for (int k0 = 0; k0 < LEAF_W; k0 += 4) {
                v2f a, b;
                a[0] = hbuf[mn * 16 + k0 + koff];
                a[1] = hbuf[mn * 16 + k0 + koff + 1];
                b[0] = w2[(k0 + koff) * OUT_W + nt * 16 + mn];
                b[1] = w2[(k0 + koff + 1) * OUT_W + nt * 16 + mn];
                o = __builtin_amdgcn_wmma_f32_16x16x4_f32(
                        false, a, false, b, (short)0, o, false, false);
            }
#pragma unroll
            for (int r = 0; r < 8; ++r) {
                const int trow = rbase + r;
                if (trow < nrows) {
                    const int s = order[base + t0 + trow];
                    out[(size_t)s * OUT_W + nt * 16 + mn] = o[r];
                }
            }
        }
        __syncthreads();   // hbuf reused by next tile
    }
}

// ---------------------------------------------------------------------------
extern "C" void kernel_launch(void* const* d_in, const int* in_sizes, int n_in,
                              void* d_out, int out_size, void* d_ws, size_t ws_size,
                              hipStream_t stream) {
    const float* x   = (const float*)d_in[0];
    const float* nw  = (const float*)d_in[1];
    const float* nb  = (const float*)d_in[2];
    const float* w1s = (const float*)d_in[3];
    const float* b1s = (const float*)d_in[4];
    const float* w2s = (const float*)d_in[5];
    float* out = (float*)d_out;

    // workspace layout (ints): leaf[8192] counts[2048] offsets[2048]
    //                          cursors[2048] order[8192]  (~90 KB)
    int* leaf    = (int*)d_ws;
    int* counts  = leaf + BATCH;
    int* offsets = counts + N_LEAVES;
    int* cursors = offsets + N_LEAVES;
    int* order   = cursors + N_LEAVES;

    fff_zero<<<(N_LEAVES + 255) / 256, 256, 0, stream>>>(counts);
    fff_route<<<(BATCH * 32) / 256, 256, 0, stream>>>(x, nw, nb, leaf, counts);
    fff_scan<<<1, 256, 0, stream>>>(counts, offsets, cursors);
    fff_scatter<<<BATCH / 256, 256, 0, stream>>>(leaf, cursors, order);
    fff_leaf_mlp<<<N_LEAVES, 32, 0, stream>>>(x, w1s, b1s, w2s,
                                              order, offsets, counts, out);
}